// Graph_Learn_74354473828597
// MI455X (gfx1250) — compile-verified
//
#include <hip/hip_runtime.h>

typedef __attribute__((ext_vector_type(2))) float v2f;
typedef __attribute__((ext_vector_type(8))) float v8f;

#define SSTR 68  // padded S-tile row stride (floats): conflict-free WMMA A reads

__global__ __launch_bounds__(256) void graph_learn_kernel(
    const float* __restrict__ x, const float* __restrict__ a,
    float* __restrict__ out)
{
  // Dynamic LDS layout (floats):
  //  xs    : 64 rows x 128 floats (one f-half), XOR-swizzled float4 groups
  //  s_lds : 32 x SSTR exp(S) tile for this block's i-rows
  //  a_lds : 256 weights
  //  denom : 32 row sums
  extern __shared__ float smem[];
  float* xs    = smem;               // 8192 floats
  float* s_lds = smem + 64 * 128;    // 2176 floats
  float* a_lds = s_lds + 32 * SSTR;  // 256 floats
  float* denom = a_lds + 256;        // 32 floats

  const int tid   = threadIdx.x;
  const int n     = blockIdx.x;   // 0..63
  const int ihalf = blockIdx.y;   // 0..1 : which 32 i-rows this block owns

  // xm = x[n, t_mid=2, :, :]  (64 x 256 fp32)
  const float4* xg4 = (const float4*)(x + (size_t)(n * 5 + 2) * (64 * 256));
  __builtin_prefetch(xg4, 0, 3);  // gfx1250 global_prefetch_b8

  if (tid < 64) ((float4*)a_lds)[tid] = ((const float4*)a)[tid];

  const int ti  = tid >> 4;              // 0..15
  const int tj  = tid & 15;              // 0..15
  const int i0  = ihalf * 32 + ti * 2;   // first of this thread's 2 i-rows
  const int j0  = tj * 4;                // first of this thread's 4 j-rows
  const int gi0 = (i0) >> 2;             // swizzle keys for xi reads
  const int gi1 = (i0 + 1) >> 2;

  float acc[2][4];
#pragma unroll
  for (int ii = 0; ii < 2; ++ii)
#pragma unroll
    for (int jj = 0; jj < 4; ++jj) acc[ii][jj] = 0.0f;

  float4* xs4 = (float4*)xs;
  const float4* a4p = (const float4*)a_lds;

  // Uniform 64-bit global base for GVS-mode async loads (lands in an SGPR pair).
  const unsigned long long gbase = (unsigned long long)(uintptr_t)xg4;

  // Two f-halves of 128 floats each keep LDS under 48KB.
  for (int h = 0; h < 2; ++h) {
    // Stage one f-half directly into LDS with the CDNA5 async mover:
    // GLOBAL_LOAD_ASYNC_TO_LDS_B128 (GVS: mem = s[base] + v[off]); the per-lane
    // LDS dest VGPR carries the XOR swizzle (group f4 ^ (row>>2)).
    // Wave-contiguous global reads (512B/wave), conflict-free LDS writes.
#pragma unroll
    for (int k = 0; k < 8; ++k) {
      int linear = tid + 256 * k;   // 0..2047
      int v  = linear >> 5;         // row 0..63
      int f4 = linear & 31;         // group within half
      unsigned int lds_byte =
          (unsigned int)(uintptr_t)(&xs4[v * 32 + (f4 ^ (v >> 2))]);
      unsigned int goff = (unsigned int)((v * 64 + h * 32 + f4) * 16);
      asm volatile("global_load_async_to_lds_b128 %0, %1, %2"
                   :
                   : "v"(lds_byte), "v"(goff), "s"(gbase)
                   : "memory");
    }
    asm volatile("s_wait_asynccnt 0x0" ::: "memory");
    __syncthreads();

    for (int c = 0; c < 32; ++c) {
      float4 av  = a4p[h * 32 + c];
      float4 xi0 = xs4[(i0    ) * 32 + (c ^ gi0)];   // broadcast reads
      float4 xi1 = xs4[(i0 + 1) * 32 + (c ^ gi1)];
#pragma unroll
      for (int jj = 0; jj < 4; ++jj) {
        float4 xj = xs4[(j0 + jj) * 32 + (c ^ tj)];  // 16 distinct bank groups
        acc[0][jj] = fmaf(av.x, fabsf(xi0.x - xj.x), acc[0][jj]);
        acc[0][jj] = fmaf(av.y, fabsf(xi0.y - xj.y), acc[0][jj]);
        acc[0][jj] = fmaf(av.z, fabsf(xi0.z - xj.z), acc[0][jj]);
        acc[0][jj] = fmaf(av.w, fabsf(xi0.w - xj.w), acc[0][jj]);
        acc[1][jj] = fmaf(av.x, fabsf(xi1.x - xj.x), acc[1][jj]);
        acc[1][jj] = fmaf(av.y, fabsf(xi1.y - xj.y), acc[1][jj]);
        acc[1][jj] = fmaf(av.z, fabsf(xi1.z - xj.z), acc[1][jj]);
        acc[1][jj] = fmaf(av.w, fabsf(xi1.w - xj.w), acc[1][jj]);
      }
    }
    __syncthreads();  // protect xs before next half is staged
  }

  // exp() and stage local S tile (32 x 64, stride SSTR)
#pragma unroll
  for (int ii = 0; ii < 2; ++ii) {
    float4 e;
    e.x = __expf(acc[ii][0]); e.y = __expf(acc[ii][1]);
    e.z = __expf(acc[ii][2]); e.w = __expf(acc[ii][3]);
    acc[ii][0] = e.x; acc[ii][1] = e.y; acc[ii][2] = e.z; acc[ii][3] = e.w;
    *(float4*)&s_lds[(ti * 2 + ii) * SSTR + j0] = e;
  }
  __syncthreads();

  // Row-sum reduction via WMMA: D = A(16x4 S-chunk) x B(4x16 ones) + C.
  // With B == ones, D[m,n] = sum_k A[m,k]; chaining 16 K-chunks covers j=0..63.
  // Result is invariant to the A K-permutation; uses documented M = lane%16.
  const int wave = tid >> 5;
  const int lane = tid & 31;
  if (wave < 2) {                       // wave-uniform -> EXEC all ones
    const int m  = lane & 15;           // A-row this lane feeds
    const int kb = (lane < 16) ? 0 : 2; // K-pair held by this lane half
    const float* arow = &s_lds[(wave * 16 + m) * SSTR + kb];
    v2f bones = {1.0f, 1.0f};
    v8f c8 = {0.f, 0.f, 0.f, 0.f, 0.f, 0.f, 0.f, 0.f};
#pragma unroll
    for (int c = 0; c < 16; ++c) {
      v2f a2;
      a2.x = arow[c * 4 + 0];
      a2.y = arow[c * 4 + 1];
      c8 = __builtin_amdgcn_wmma_f32_16x16x4_f32(
          /*neg_a=*/false, a2, /*neg_b=*/false, bones,
          /*c_mod=*/(short)0, c8, /*reuse_a=*/false, /*reuse_b=*/false);
    }
    // D layout: VGPR r -> M=r (lanes 0-15) / M=8+r (lanes 16-31); take N=0 lanes.
    if (m == 0) {
      const int rbase = wave * 16 + ((lane >> 4) << 3);
#pragma unroll
      for (int r = 0; r < 8; ++r) denom[rbase + r] = c8[r];
    }
  }
  __syncthreads();

  // out[n,i,j] = S[i,j] / denom[i]  (== reference's axis-1 sum by exact symmetry)
  float4* out4 = (float4*)out;
#pragma unroll
  for (int ii = 0; ii < 2; ++ii) {
    const int i = i0 + ii;
    const float ds = denom[ti * 2 + ii];
    float4 o;
    o.x = acc[ii][0] / ds; o.y = acc[ii][1] / ds;
    o.z = acc[ii][2] / ds; o.w = acc[ii][3] / ds;
    out4[(n * 64 + i) * 16 + tj] = o;
  }
}

extern "C" void kernel_launch(void* const* d_in, const int* in_sizes, int n_in,
                              void* d_out, int out_size, void* d_ws, size_t ws_size,
                              hipStream_t stream) {
  const float* x = (const float*)d_in[0];  // (64,5,64,256) fp32
  const float* a = (const float*)d_in[1];  // (256,1) fp32
  float* out = (float*)d_out;              // (64,64,64) fp32
  dim3 grid(64, 2), block(256);
  size_t lds_bytes = (size_t)(64 * 128 + 32 * SSTR + 256 + 32) * sizeof(float);
  graph_learn_kernel<<<grid, block, lds_bytes, stream>>>(x, a, out);
}